// HyperedgeAggregator_11218454577211
// MI455X (gfx1250) — compile-verified
//
#include <hip/hip_runtime.h>
#include <hip/hip_bf16.h>

typedef __attribute__((ext_vector_type(2))) float v2f;
typedef __attribute__((ext_vector_type(4))) float v4f;
typedef __attribute__((ext_vector_type(8))) float v8f;

#define DIM 256
#define GROUP_TOTAL 32  // 4 groups * 8 nodes; equal-size nested means == flat mean

// ---------------------------------------------------------------------------
// Kernel 1: H = relu(X @ W^T + b), fp32 WMMA (V_WMMA_F32_16X16X4_F32).
// One wave32 computes one 16x16 output tile. K advances 4 per WMMA.
//
// A-frag (16x4, MxK) per ISA layout: lanes 0-15 -> M=lane, K={0,1};
//                                    lanes 16-31 -> M=lane-16, K={2,3}.
// B-frag (4x16, KxN): lanes 0-15 -> N=lane, K={0,1}; lanes 16-31 -> K={2,3}.
// B = W^T, i.e. B[k][n] = W[n][k], so each lane loads 2 consecutive floats
// from row (n0+lane&15) of row-major W — a clean global_load_b64.
// ---------------------------------------------------------------------------
__global__ __launch_bounds__(256) void hyperedge_gemm_relu_wmma(
    const float* __restrict__ X,   // [nNodes, DIM]
    const float* __restrict__ W,   // [DIM, DIM] row-major
    const float* __restrict__ b,   // [DIM]
    float* __restrict__ H,         // [nNodes, DIM]
    int mTiles)
{
    const int lane = threadIdx.x & 31;
    const int wave = threadIdx.x >> 5;
    const int tile = blockIdx.x * 8 + wave;   // wave-uniform
    const int nTile = tile & 15;              // DIM/16 == 16 column tiles
    const int mTile = tile >> 4;
    if (mTile >= mTiles) return;              // uniform branch: EXEC stays all-1s

    const int m0   = mTile * 16;
    const int n0   = nTile * 16;
    const int half = lane >> 4;               // 0: K={0,1}; 1: K={2,3}
    const int l15  = lane & 15;

    const float* aPtr = X + (size_t)(m0 + l15) * DIM + half * 2;
    const float* bPtr = W + (size_t)(n0 + l15) * DIM + half * 2;

    v8f acc = {};
#pragma unroll 4
    for (int k = 0; k < DIM; k += 4) {
        v2f aFrag = *(const v2f*)(aPtr + k);
        v2f bFrag = *(const v2f*)(bPtr + k);
        // 8 args: (neg_a, A, neg_b, B, c_mod, C, reuse_a, reuse_b)
        acc = __builtin_amdgcn_wmma_f32_16x16x4_f32(
            false, aFrag, false, bFrag, (short)0, acc, false, false);
    }

    // C/D layout: VGPR r, lanes 0-15 -> row m0+r, lanes 16-31 -> row m0+8+r;
    // column n0 + (lane&15) for both halves.
    const float bias = b[n0 + l15];
    float* out = H + (size_t)(m0 + half * 8) * DIM + n0 + l15;
#pragma unroll
    for (int r = 0; r < 8; ++r) {
        float v = acc[r] + bias;
        v = v > 0.0f ? v : 0.0f;               // ReLU
        out[(size_t)r * DIM] = v;
    }
}

// ---------------------------------------------------------------------------
// Kernel 2: out[e] = mean over 32 gathered rows of H.
// 64 lanes per edge, each owning a float4 slice of the 256-dim row.
// Gathered rows stream from L2 (H = 102.4 MB < 192 MB global L2).
// ---------------------------------------------------------------------------
__global__ __launch_bounds__(256) void hyperedge_gather_mean(
    const float* __restrict__ H,    // [nNodes, DIM]
    const int*   __restrict__ idx,  // [nEdges, 32] flattened
    float* __restrict__ out,        // [nEdges, DIM]
    int nEdges)
{
    const int sub = threadIdx.x >> 6;   // edge slot within block (0..3)
    const int t   = threadIdx.x & 63;   // float4 slot within row (0..63)
    const int e   = blockIdx.x * 4 + sub;
    if (e >= nEdges) return;

    const int* ePtr = idx + (size_t)e * GROUP_TOTAL;
    v4f sum = {};
#pragma unroll 8
    for (int j = 0; j < GROUP_TOTAL; ++j) {
        const int node = ePtr[j];
        v4f v = *(const v4f*)(H + (size_t)node * DIM + t * 4);
        sum += v;
    }
    sum *= (1.0f / (float)GROUP_TOTAL);   // mean over groups(4) x group_size(8)
    *(v4f*)(out + (size_t)e * DIM + t * 4) = sum;
}

extern "C" void kernel_launch(void* const* d_in, const int* in_sizes, int n_in,
                              void* d_out, int out_size, void* d_ws, size_t ws_size,
                              hipStream_t stream) {
    // setup_inputs() order: node_embeddings, subset_indices, W, b
    const float* X   = (const float*)d_in[0];
    const int*   idx = (const int*)  d_in[1];
    const float* W   = (const float*)d_in[2];
    const float* b   = (const float*)d_in[3];
    float* out = (float*)d_out;
    float* H   = (float*)d_ws;     // [nNodes, DIM] fp32 scratch (~102.4 MB)

    const int nNodes = in_sizes[0] / DIM;          // 100000
    const int nEdges = in_sizes[1] / GROUP_TOTAL;  // 20000

    const int mTiles     = (nNodes + 15) / 16;     // 6250 (exact)
    const int totalTiles = mTiles * (DIM / 16);    // 100000 wave-tiles
    const int blocks1    = (totalTiles + 7) / 8;   // 8 waves per 256-thread block

    hyperedge_gemm_relu_wmma<<<blocks1, 256, 0, stream>>>(X, W, b, H, mTiles);

    const int blocks2 = (nEdges + 3) / 4;          // 4 edges per 256-thread block
    hyperedge_gather_mean<<<blocks2, 256, 0, stream>>>(H, idx, out, nEdges);
}